// BTM_56358560858814
// MI455X (gfx1250) — compile-verified
//
#include <hip/hip_runtime.h>
#include <math.h>

// Problem constants (from reference)
#define RADIUS  5
#define NIMG    8
#define HH      352
#define WW      352
#define TILE    32
#define LT      (TILE + 2*RADIUS)        // 42
#define NTX     (WW / TILE)              // 11
#define NTY     (HH / TILE)              // 11
#define NBLOCKS (NIMG * NTX * NTY)       // 968

// -ALPHA * log2(e):  exp(-200*s) == exp2(K2*s)
#define K2 (-288.53900817779268f)

typedef float v2f __attribute__((ext_vector_type(2)));
typedef float v8f __attribute__((ext_vector_type(8)));

// ---------------------------------------------------------------------------
// Stage 1: fused boundary-mask + 11x11 bilateral stencil, per-block partials.
// One block = one 32x32 output tile of one image. 256 threads, each owns 4
// horizontally adjacent pixels (register-reuse of the sliding window).
// ---------------------------------------------------------------------------
__global__ __launch_bounds__(256)
void BTM_stencil_kernel(const float* __restrict__ pred,
                        const float* __restrict__ feat,
                        float* __restrict__ ws)
{
    __shared__ float4 tile[LT][LT];     // (r,g,b,sal); OOB = (0,0,0,-1)
    __shared__ float  red[512];

    const int tx = blockIdx.x, ty = blockIdx.y, n = blockIdx.z;
    const int x0 = tx * TILE, y0 = ty * TILE;
    const int tid = threadIdx.x;

    const float* predn = pred + (size_t)n * HH * WW;
    const float* featr = feat + (size_t)(n*3 + 0) * HH * WW;
    const float* featg = feat + (size_t)(n*3 + 1) * HH * WW;
    const float* featb = feat + (size_t)(n*3 + 2) * HH * WW;

    // ---- stage tile (+halo) into LDS, zero-pad rgb / sentinel sal at OOB ----
    for (int s = tid; s < LT*LT; s += 256) {
        const int ly = s / LT, lx = s - ly * LT;
        const int gy = y0 + ly - RADIUS, gx = x0 + lx - RADIUS;
        float4 v;
        if (gy >= 0 && gy < HH && gx >= 0 && gx < WW) {
            const int idx = gy * WW + gx;
            v.x = featr[idx]; v.y = featg[idx]; v.z = featb[idx];
            v.w = predn[idx];
        } else {
            v = make_float4(0.f, 0.f, 0.f, -1.f);   // -1 marks OOB (pred >= 0)
        }
        tile[ly][lx] = v;
    }
    __syncthreads();

    const int oy  = tid >> 3;          // 0..31   (tile row)
    const int ox0 = (tid & 7) * 4;     // 0..28   (first of 4 tile cols)
    const int cy  = oy + RADIUS;

    // ---- boundary mask: clipped-window 5x5 max/min over labels ----
    // dil: OOB sentinel -> label 0 (neutral for max)
    // ero: OOB sentinel -> label 1 (neutral for min == clipped window)
    bool dil[4] = {false,false,false,false};
    bool ero[4] = {true, true, true, true};
    for (int wy = 0; wy < 5; ++wy) {
        const int ly = cy - 2 + wy;
        float sv[8];
        #pragma unroll
        for (int k = 0; k < 8; ++k) sv[k] = tile[ly][ox0 + 3 + k].w;
        #pragma unroll
        for (int j = 0; j < 4; ++j) {
            #pragma unroll
            for (int k = 0; k < 5; ++k) {
                const float s = sv[j + k];
                const bool lab1 = (s > 0.5f);
                dil[j] = dil[j] || lab1;
                ero[j] = ero[j] && (lab1 || (s < 0.f));
            }
        }
    }
    float maskf[4]; int anymask = 0;
    #pragma unroll
    for (int j = 0; j < 4; ++j) {
        const int m = (dil[j] ? 1 : 0) - (ero[j] ? 1 : 0);   // in {0,1}
        maskf[j] = (float)m;
        anymask |= m;
    }

    float local_loss = 0.f;
    float local_mask = maskf[0] + maskf[1] + maskf[2] + maskf[3];

    // ---- 11x11 bilateral stencil (only where the mask can contribute) ----
    if (anymask) {
        float4 c[4];
        #pragma unroll
        for (int j = 0; j < 4; ++j) c[j] = tile[cy][ox0 + RADIUS + j];

        float acc[4] = {0.f, 0.f, 0.f, 0.f};
        for (int dy = 0; dy < 11; ++dy) {
            const int ly = cy + dy - RADIUS;
            float4 w[14];                         // shared by 4 centers x 11 dx
            #pragma unroll
            for (int k = 0; k < 14; ++k) w[k] = tile[ly][ox0 + k];
            #pragma unroll
            for (int j = 0; j < 4; ++j) {
                float lm = 0.f;
                #pragma unroll
                for (int dx = 0; dx < 11; ++dx) {
                    const float4 t = w[j + dx];
                    const float dr = t.x - c[j].x;
                    const float dg = t.y - c[j].y;
                    const float db = t.z - c[j].z;
                    float s2 = dr * dr;
                    s2 = fmaf(dg, dg, s2);
                    s2 = fmaf(db, db, s2);
                    const float wgt  = __builtin_amdgcn_exp2f(K2 * s2);
                    const float dsal = fabsf(fmaxf(t.w, 0.f) - c[j].w); // zero-pad semantics
                    lm = fmaf(wgt, dsal, lm);
                }
                acc[j] += lm;
            }
        }
        #pragma unroll
        for (int j = 0; j < 4; ++j) local_loss = fmaf(maskf[j], acc[j], local_loss);
    }

    // ---- deterministic block reduction ----
    red[tid]       = local_loss;
    red[256 + tid] = local_mask;
    __syncthreads();
    for (int off = 128; off > 0; off >>= 1) {
        if (tid < off) {
            red[tid]       += red[tid + off];
            red[256 + tid] += red[256 + tid + off];
        }
        __syncthreads();
    }
    if (tid == 0) {
        const int b = (n * NTY + ty) * NTX + tx;
        ws[2*b]     = red[0];
        ws[2*b + 1] = red[256];
    }
}

// ---------------------------------------------------------------------------
// Stage 2: single-wave deterministic reduction of per-block partials.
// Cross-lane sum done with V_WMMA_F32_16X16X4_F32 (ones-matrix row-sum trick,
// full fp32, layout-robust since B is all-ones).
// ---------------------------------------------------------------------------
__global__ __launch_bounds__(32)
void BTM_reduce_kernel(const float* __restrict__ ws, float* __restrict__ out,
                       int nblocks)
{
    const int lane = threadIdx.x;      // wave32
    float lsum = 0.f, msum = 0.f;
    for (int i = lane; i < nblocks; i += 32) {
        lsum += ws[2*i];
        msum += ws[2*i + 1];
    }

    v2f b;  b.x = 1.f; b.y = 1.f;          // all-ones B (4x16)
    v8f c = {0.f,0.f,0.f,0.f,0.f,0.f,0.f,0.f};

    v2f a;  a.x = lsum; a.y = 0.f;         // each lane contributes one A slot
    v8f dl = __builtin_amdgcn_wmma_f32_16x16x4_f32(
        false, a, false, b, (short)0, c, false, false);

    a.x = msum;
    v8f dm = __builtin_amdgcn_wmma_f32_16x16x4_f32(
        false, a, false, b, (short)0, c, false, false);

    // D[m][n] = sum over K of A-row m; sum the 8 rows held in-lane, then the
    // two lane-halves of column 0 (lanes 0 and 16) give the wave total.
    float Sl = dl[0]+dl[1]+dl[2]+dl[3]+dl[4]+dl[5]+dl[6]+dl[7];
    float Sm = dm[0]+dm[1]+dm[2]+dm[3]+dm[4]+dm[5]+dm[6]+dm[7];
    const float tl = __shfl(Sl, 0, 32) + __shfl(Sl, 16, 32);
    const float tm = __shfl(Sm, 0, 32) + __shfl(Sm, 16, 32);

    if (lane == 0) out[0] = tl / (tm + 1e-6f);
}

// ---------------------------------------------------------------------------
extern "C" void kernel_launch(void* const* d_in, const int* in_sizes, int n_in,
                              void* d_out, int out_size, void* d_ws, size_t ws_size,
                              hipStream_t stream)
{
    const float* pred = (const float*)d_in[0];   // (8,1,352,352) f32
    const float* feat = (const float*)d_in[1];   // (8,3,352,352) f32
    float* out = (float*)d_out;                  // scalar f32
    float* ws  = (float*)d_ws;                   // >= NBLOCKS*2 floats

    dim3 grid(NTX, NTY, NIMG);
    BTM_stencil_kernel<<<grid, 256, 0, stream>>>(pred, feat, ws);
    BTM_reduce_kernel<<<1, 32, 0, stream>>>(ws, out, NBLOCKS);
}